// GCN_62139586839006
// MI455X (gfx1250) — compile-verified
//
#include <hip/hip_runtime.h>

typedef __attribute__((ext_vector_type(2))) float v2f;
typedef __attribute__((ext_vector_type(8))) float v8f;

#define GCN_N_GRAPHS 64

// ---------------- utility kernels ----------------

__global__ void gcn_fill_zero(float* __restrict__ p, int n) {
    int i = blockIdx.x * blockDim.x + threadIdx.x;
    if (i < n) p[i] = 0.0f;
}

__global__ void gcn_degree(const int* __restrict__ src, const int* __restrict__ dst,
                           float* __restrict__ deg_out, float* __restrict__ deg_in, int E) {
    int e = blockIdx.x * blockDim.x + threadIdx.x;
    if (e < E) {
        unsafeAtomicAdd(&deg_out[src[e]], 1.0f);
        unsafeAtomicAdd(&deg_in[dst[e]], 1.0f);
    }
}

__global__ void gcn_norm_finalize(float* __restrict__ buf, int n) {
    int i = blockIdx.x * blockDim.x + threadIdx.x;
    if (i < n) {
        float d = buf[i];
        d = (d < 1.0f) ? 1.0f : d;
        buf[i] = 1.0f / sqrtf(d);
    }
}

__global__ void gcn_count(const int* __restrict__ gid, float* __restrict__ counts, int n) {
    int i = blockIdx.x * blockDim.x + threadIdx.x;
    if (i < n) unsafeAtomicAdd(&counts[gid[i]], 1.0f);
}

__global__ void gcn_final(const float* __restrict__ pooled, const float* __restrict__ counts,
                          float* __restrict__ out) {
    int i = blockIdx.x * blockDim.x + threadIdx.x;  // 64 graphs * 64 feats
    if (i < GCN_N_GRAPHS * 64) {
        float c = counts[i >> 6];
        c = (c < 1.0f) ? 1.0f : c;
        out[i] = pooled[i] / c;
    }
}

// ---------------- SpMM: agg[dst] += x[src] * out_norm[src] ----------------
// F/4 threads per edge, each moves one float4: coalesced gather, 4 f32 atomics.

template <int F>
__global__ void gcn_spmm(const float* __restrict__ x, const float* __restrict__ out_norm,
                         const int* __restrict__ src, const int* __restrict__ dst,
                         float* __restrict__ agg, int E) {
    const int TPE = F / 4;
    int gtid = blockIdx.x * blockDim.x + threadIdx.x;
    int e = gtid / TPE;
    if (e >= E) return;
    int f4 = (gtid - e * TPE) * 4;
    int s = src[e];
    int d = dst[e];
    float sc = out_norm[s];
    const float4 v = *reinterpret_cast<const float4*>(x + (size_t)s * F + f4);
    float* base = agg + (size_t)d * F + f4;
    unsafeAtomicAdd(base + 0, v.x * sc);
    unsafeAtomicAdd(base + 1, v.y * sc);
    unsafeAtomicAdd(base + 2, v.z * sc);
    unsafeAtomicAdd(base + 3, v.w * sc);
}

// ---------------- GEMM via V_WMMA_F32_16X16X4_F32 ----------------
// out = act( (A * in_norm[row]) @ W + bias ), one 16x16 tile per wave32.
// POOL epilogue scatters rows into per-graph pooled sums instead of storing.

template <bool RELU, bool POOL>
__global__ void gcn_gemm_wmma(const float* __restrict__ A, const float* __restrict__ in_norm,
                              const float* __restrict__ W, const float* __restrict__ bias,
                              float* __restrict__ out, int M, int K, int Nw,
                              const int* __restrict__ graph_ids, float* __restrict__ pooled) {
    int wave = blockIdx.x * (blockDim.x >> 5) + (threadIdx.x >> 5);
    int tiles_n = Nw >> 4;
    int total = (M >> 4) * tiles_n;
    if (wave >= total) return;                 // wave-uniform: EXEC stays all-ones
    int mt = wave / tiles_n;
    int nt = wave - mt * tiles_n;
    int row0 = mt << 4;
    int col0 = nt << 4;

    int lane  = threadIdx.x & 31;
    int idx15 = lane & 15;
    int koff  = (lane >> 4) << 1;              // 0 for lanes 0-15, 2 for lanes 16-31

    // A fragment source: row M = idx15, K slots {koff, koff+1} (ISA 16x4 f32 layout)
    int rowA = row0 + idx15;
    float scale = in_norm[rowA];
    const float* arow = A + (size_t)rowA * K + koff;
    // B fragment source: col N = idx15, K slots {koff, koff+1}
    const float* wcol = W + col0 + idx15;

    v8f acc = {};
    for (int k = 0; k < K; k += 4) {
        float2 av = *reinterpret_cast<const float2*>(arow + k);
        v2f af;
        af.x = av.x * scale;
        af.y = av.y * scale;
        v2f bf;
        bf.x = wcol[(size_t)(k + koff) * Nw];
        bf.y = wcol[(size_t)(k + koff + 1) * Nw];
        acc = __builtin_amdgcn_wmma_f32_16x16x4_f32(
            /*neg_a=*/false, af, /*neg_b=*/false, bf,
            /*c_mod=*/(short)0, acc, /*reuse_a=*/false, /*reuse_b=*/false);
    }

    // C/D layout: VGPR r holds M = r (lanes 0-15) or M = 8 + r (lanes 16-31); N = idx15
    int col = col0 + idx15;
    float bv = bias[col];
    int mbase = (lane >> 4) << 3;
#pragma unroll
    for (int r = 0; r < 8; ++r) {
        int m = row0 + mbase + r;
        float v = acc[r] + bv;
        if (RELU) v = (v > 0.0f) ? v : 0.0f;
        if (POOL) {
            int g = graph_ids[m];
            unsafeAtomicAdd(&pooled[g * Nw + col], v);
        } else {
            out[(size_t)m * Nw + col] = v;
        }
    }
}

// ---------------- driver ----------------

static inline int cdiv_l(long a, long b) { return (int)((a + b - 1) / b); }

extern "C" void kernel_launch(void* const* d_in, const int* in_sizes, int n_in,
                              void* d_out, int out_size, void* d_ws, size_t ws_size,
                              hipStream_t stream) {
    const float* emb = (const float*)d_in[0];
    const float* W1  = (const float*)d_in[1];
    const float* b1  = (const float*)d_in[2];
    const float* W2  = (const float*)d_in[3];
    const float* b2  = (const float*)d_in[4];
    const float* W3  = (const float*)d_in[5];
    const float* b3  = (const float*)d_in[6];
    const int* src = (const int*)d_in[7];
    const int* dst = (const int*)d_in[8];
    const int* gid = (const int*)d_in[9];

    const int E = in_sizes[7];
    const int N = in_sizes[9];   // 100000, multiple of 16

    // workspace layout (floats)
    float* ws       = (float*)d_ws;
    float* out_norm = ws;                         // N
    float* in_norm  = out_norm + N;               // N
    float* pooled   = in_norm + N;                // 64*64
    float* counts   = pooled + GCN_N_GRAPHS * 64; // 64
    float* agg      = counts + GCN_N_GRAPHS;      // N*128  (16B-aligned for N=100000)
    float* h        = agg + (size_t)N * 128;      // N*128

    const int T = 256;

    // zero norms + pooled + counts (contiguous prefix)
    int zc = 2 * N + GCN_N_GRAPHS * 64 + GCN_N_GRAPHS;
    gcn_fill_zero<<<cdiv_l(zc, T), T, 0, stream>>>(ws, zc);

    // degrees -> D^{-1/2} norms; per-graph node counts
    gcn_degree<<<cdiv_l(E, T), T, 0, stream>>>(src, dst, out_norm, in_norm, E);
    gcn_norm_finalize<<<cdiv_l(2L * N, T), T, 0, stream>>>(out_norm, 2 * N);
    gcn_count<<<cdiv_l(N, T), T, 0, stream>>>(gid, counts, N);

    // ---- Layer 1: emb(128) -> h(128), ReLU ----
    gcn_fill_zero<<<cdiv_l((long)N * 128, T), T, 0, stream>>>(agg, N * 128);
    gcn_spmm<128><<<cdiv_l((long)E * 32, T), T, 0, stream>>>(emb, out_norm, src, dst, agg, E);
    {
        int tiles = (N / 16) * (128 / 16);
        gcn_gemm_wmma<true, false><<<cdiv_l(tiles, 4), 128, 0, stream>>>(
            agg, in_norm, W1, b1, h, N, 128, 128, nullptr, nullptr);
    }

    // ---- Layer 2: h(128) -> h(64), ReLU ----
    gcn_fill_zero<<<cdiv_l((long)N * 128, T), T, 0, stream>>>(agg, N * 128);
    gcn_spmm<128><<<cdiv_l((long)E * 32, T), T, 0, stream>>>(h, out_norm, src, dst, agg, E);
    {
        int tiles = (N / 16) * (64 / 16);
        gcn_gemm_wmma<true, false><<<cdiv_l(tiles, 4), 128, 0, stream>>>(
            agg, in_norm, W2, b2, h, N, 128, 64, nullptr, nullptr);
    }

    // ---- Layer 3: h(64) -> pooled sums (fused pooling epilogue) ----
    gcn_fill_zero<<<cdiv_l((long)N * 64, T), T, 0, stream>>>(agg, N * 64);
    gcn_spmm<64><<<cdiv_l((long)E * 16, T), T, 0, stream>>>(h, out_norm, src, dst, agg, E);
    {
        int tiles = (N / 16) * (64 / 16);
        gcn_gemm_wmma<false, true><<<cdiv_l(tiles, 4), 128, 0, stream>>>(
            agg, in_norm, W3, b3, nullptr, N, 64, 64, gid, pooled);
    }

    // ---- mean over nodes per graph ----
    gcn_final<<<cdiv_l(GCN_N_GRAPHS * 64, T), T, 0, stream>>>(pooled, counts, (float*)d_out);
}